// FAPELoss_82368882802877
// MI455X (gfx1250) — compile-verified
//
#include <hip/hip_runtime.h>

// ---------------------------------------------------------------------------
// FAPE loss for MI455X (gfx1250).
// Pairwise stage rewritten as a rank-20 factorized bilinear form so the
// 384 x 5376 x K pairwise squared-distance computation maps onto
// V_WMMA_F32_16X16X4_F32 (wave32, 16x16 f32 tiles, K-chunks of 4).
// Epilogue uses raw V_SQRT_F32 (operand is >= 1e-8, so the libm denormal
// guard sequence is unnecessary) and factors the point mask out of the
// 8-row accumulation.
// ---------------------------------------------------------------------------

typedef float v2f __attribute__((ext_vector_type(2)));
typedef float v8f __attribute__((ext_vector_type(8)));

#define EPSF   1e-8f
#define DCLAMP 10.0f

#define Bz  2
#define Nn  384
#define Aa  14
#define NA  (Nn * Aa)        // 5376 points per batch
#define KF  20               // 17 feature dims padded to 20 (5 chunks of K=4)
#define FT  (Nn / 16)        // 24 frame tiles
#define PT  (NA / 16)        // 336 point tiles
#define SPLIT 7
#define PT_PER_BLOCK (PT / SPLIT)   // 48 point tiles per block

// ---------------------------------------------------------------------------
// Gram-Schmidt frame from backbone atoms (N, CA, C). e[k][comp] = basis vec k.
// ---------------------------------------------------------------------------
__device__ inline void build_frame(const float* __restrict__ at,
                                   float e[3][3], float t[3]) {
    float nx = at[0], ny = at[1], nz = at[2];
    float cax = at[3], cay = at[4], caz = at[5];
    float cx = at[6], cy = at[7], cz = at[8];
    float v1x = cx - cax, v1y = cy - cay, v1z = cz - caz;
    float v2x = nx - cax, v2y = ny - cay, v2z = nz - caz;
    float inv1 = rsqrtf(v1x * v1x + v1y * v1y + v1z * v1z + EPSF);
    float e1x = v1x * inv1, e1y = v1y * inv1, e1z = v1z * inv1;
    float dp = e1x * v2x + e1y * v2y + e1z * v2z;
    float u2x = v2x - e1x * dp, u2y = v2y - e1y * dp, u2z = v2z - e1z * dp;
    float inv2 = rsqrtf(u2x * u2x + u2y * u2y + u2z * u2z + EPSF);
    float e2x = u2x * inv2, e2y = u2y * inv2, e2z = u2z * inv2;
    float e3x = e1y * e2z - e1z * e2y;
    float e3y = e1z * e2x - e1x * e2z;
    float e3z = e1x * e2y - e1y * e2x;
    e[0][0] = e1x; e[0][1] = e1y; e[0][2] = e1z;
    e[1][0] = e2x; e[1][1] = e2y; e[1][2] = e2z;
    e[2][0] = e3x; e[2][1] = e3y; e[2][2] = e3z;
    t[0] = cax; t[1] = cay; t[2] = caz;
}

// ---------------------------------------------------------------------------
// Kernel 1: per-frame feature vectors F[b*N][KF] + residue mask.
//   ||d_ij||^2 = F_i . P_j  with
//   F = [1, |tp|^2+|tt|^2-2 tp^T M tt, -2tp+2M tt (3), -2tt+2M^T tp (3),
//        -2M (9), 0,0,0],   M = R_p R_t^T
// ---------------------------------------------------------------------------
__global__ void fape_frames(const float* __restrict__ pred,
                            const float* __restrict__ truec,
                            const unsigned char* __restrict__ cmask,
                            float* __restrict__ F,
                            float* __restrict__ fmask) {
    int i = blockIdx.x * blockDim.x + threadIdx.x;
    if (i >= Bz * Nn) return;
    const float* pp = pred  + (size_t)i * Aa * 3;
    const float* tq = truec + (size_t)i * Aa * 3;

    float ep[3][3], et[3][3], tp[3], tt[3];
    build_frame(pp, ep, tp);
    build_frame(tq, et, tt);

    float M[3][3];
#pragma unroll
    for (int a = 0; a < 3; a++)
#pragma unroll
        for (int b = 0; b < 3; b++)
            M[a][b] = ep[0][a] * et[0][b] + ep[1][a] * et[1][b] + ep[2][a] * et[2][b];

    float Mtt[3], MTtp[3];
#pragma unroll
    for (int a = 0; a < 3; a++)
        Mtt[a] = M[a][0] * tt[0] + M[a][1] * tt[1] + M[a][2] * tt[2];
#pragma unroll
    for (int b = 0; b < 3; b++)
        MTtp[b] = M[0][b] * tp[0] + M[1][b] * tp[1] + M[2][b] * tp[2];

    float tpMtt = tp[0] * Mtt[0] + tp[1] * Mtt[1] + tp[2] * Mtt[2];
    float tp2 = tp[0] * tp[0] + tp[1] * tp[1] + tp[2] * tp[2];
    float tt2 = tt[0] * tt[0] + tt[1] * tt[1] + tt[2] * tt[2];

    float* Fo = F + (size_t)i * KF;
    Fo[0] = 1.0f;
    Fo[1] = tp2 + tt2 - 2.0f * tpMtt;
#pragma unroll
    for (int a = 0; a < 3; a++) Fo[2 + a] = -2.0f * tp[a] + 2.0f * Mtt[a];
#pragma unroll
    for (int b = 0; b < 3; b++) Fo[5 + b] = -2.0f * tt[b] + 2.0f * MTtp[b];
#pragma unroll
    for (int a = 0; a < 3; a++)
#pragma unroll
        for (int b = 0; b < 3; b++) Fo[8 + 3 * a + b] = -2.0f * M[a][b];
    Fo[17] = 0.0f; Fo[18] = 0.0f; Fo[19] = 0.0f;

    const unsigned char* cm = cmask + (size_t)i * Aa;
    int any = 0;
#pragma unroll
    for (int a = 0; a < Aa; a++) any |= cm[a];
    fmask[i] = any ? 1.0f : 0.0f;
}

// ---------------------------------------------------------------------------
// Kernel 2: per-point feature vectors P[b*NA][KF] + point mask.
//   P = [|p|^2+|q|^2, 1, p (3), q (3), outer(p,q) (9), 0,0,0]
// ---------------------------------------------------------------------------
__global__ void fape_points(const float* __restrict__ pred,
                            const float* __restrict__ truec,
                            const unsigned char* __restrict__ cmask,
                            float* __restrict__ P,
                            float* __restrict__ pmask) {
    int j = blockIdx.x * blockDim.x + threadIdx.x;
    if (j >= Bz * NA) return;
    float p[3], q[3];
#pragma unroll
    for (int c = 0; c < 3; c++) { p[c] = pred[3 * (size_t)j + c]; q[c] = truec[3 * (size_t)j + c]; }
    float* Po = P + (size_t)j * KF;
    Po[0] = p[0] * p[0] + p[1] * p[1] + p[2] * p[2] +
            q[0] * q[0] + q[1] * q[1] + q[2] * q[2];
    Po[1] = 1.0f;
#pragma unroll
    for (int c = 0; c < 3; c++) { Po[2 + c] = p[c]; Po[5 + c] = q[c]; }
#pragma unroll
    for (int a = 0; a < 3; a++)
#pragma unroll
        for (int b = 0; b < 3; b++) Po[8 + 3 * a + b] = p[a] * q[b];
    Po[17] = 0.0f; Po[18] = 0.0f; Po[19] = 0.0f;
    pmask[j] = cmask[j] ? 1.0f : 0.0f;
}

// ---------------------------------------------------------------------------
// Kernel 3: WMMA tile kernel. One wave per 16x16 (frames x points) tile.
// Block = 8 waves pinned to one frame tile: A fragments loaded once,
// reused over 6 point tiles per wave. D^2 tile -> sqrt/clamp/mask epilogue
// -> wave32 shfl reduce -> one global_atomic_add_f32 per wave.
// ---------------------------------------------------------------------------
__global__ void __launch_bounds__(256)
fape_wmma(const float* __restrict__ F, const float* __restrict__ P,
          const float* __restrict__ fmask, const float* __restrict__ pmask,
          float* __restrict__ acc) {
    int blk   = blockIdx.x;               // 0 .. Bz*FT*SPLIT-1
    int split = blk % SPLIT;
    int ft    = (blk / SPLIT) % FT;
    int bb    = blk / (SPLIT * FT);
    int wave  = threadIdx.x >> 5;
    int lane  = threadIdx.x & 31;
    int mrow  = lane & 15;                // M (for A) / N (for B) index
    int khalf = lane >> 4;                // K-half select (ISA 16x4 f32 layout)

    int frameBase = bb * Nn + ft * 16;
    const float* Fb = F + (size_t)frameBase * KF;

    // A fragments: lane<16 holds K=4k+0/1 in v0/v1, lane>=16 holds K=4k+2/3.
    v2f afrag[5];
#pragma unroll
    for (int kc = 0; kc < 5; kc++) {
        const float* fr = Fb + mrow * KF + kc * 4 + khalf * 2;
        afrag[kc].x = fr[0];
        afrag[kc].y = fr[1];
    }
    // Frame masks for the 8 accumulator rows this lane owns (M = r + 8*khalf).
    float fm[8];
#pragma unroll
    for (int r = 0; r < 8; r++) fm[r] = fmask[frameBase + r + 8 * khalf];

    float lsum = 0.0f;
    int ptBase = split * PT_PER_BLOCK;
    for (int pt = ptBase + wave; pt < ptBase + PT_PER_BLOCK; pt += 8) {
        int pointBase = bb * NA + pt * 16;
        const float* Pb = P + (size_t)pointBase * KF;
        v8f c = {0.f, 0.f, 0.f, 0.f, 0.f, 0.f, 0.f, 0.f};
#pragma unroll
        for (int kc = 0; kc < 5; kc++) {
            const float* pr = Pb + mrow * KF + kc * 4 + khalf * 2;
            v2f bfrag;
            bfrag.x = pr[0];
            bfrag.y = pr[1];
            // D = A(16x4) x B(4x16) + C, f32 matrix pipe
            c = __builtin_amdgcn_wmma_f32_16x16x4_f32(
                    false, afrag[kc], false, bfrag, (short)0, c, false, false);
        }
        float pm = pmask[pointBase + mrow];
        float tsum = 0.0f;
#pragma unroll
        for (int r = 0; r < 8; r++) {
            // operand >= EPSF = 1e-8 (normal range): raw V_SQRT_F32 is safe,
            // avoids libm's denormal-guard expansion (cmp/cndmask/mul chains).
            float d2 = fmaxf(c[r], 0.0f) + EPSF;   // v_max_num_f32 + v_add
            float d  = __builtin_amdgcn_sqrtf(d2); // v_sqrt_f32
            d = fminf(d, DCLAMP);                  // v_min_num_f32
            tsum = fmaf(fm[r], d, tsum);           // v_fma_f32
        }
        lsum = fmaf(pm, tsum, lsum);               // point mask factored out
    }
    // wave32 butterfly reduction
#pragma unroll
    for (int off = 16; off > 0; off >>= 1)
        lsum += __shfl_xor(lsum, off, 32);
    if (lane == 0) atomicAdd(&acc[bb], lsum);
}

// ---------------------------------------------------------------------------
// Kernel 4: finalize. Mask count factorizes: sum(mask) = (sum fmask)(sum pmask).
// ---------------------------------------------------------------------------
__global__ void fape_finalize(const float* __restrict__ fmask,
                              const float* __restrict__ pmask,
                              const float* __restrict__ acc,
                              float* __restrict__ out) {
    __shared__ float red[4][256];
    int t = threadIdx.x;
    float fs0 = 0.f, fs1 = 0.f, ps0 = 0.f, ps1 = 0.f;
    for (int i = t; i < Nn; i += 256) { fs0 += fmask[i]; fs1 += fmask[Nn + i]; }
    for (int j = t; j < NA; j += 256) { ps0 += pmask[j]; ps1 += pmask[NA + j]; }
    red[0][t] = fs0; red[1][t] = fs1; red[2][t] = ps0; red[3][t] = ps1;
    __syncthreads();
    for (int s = 128; s > 0; s >>= 1) {
        if (t < s) {
#pragma unroll
            for (int k = 0; k < 4; k++) red[k][t] += red[k][t + s];
        }
        __syncthreads();
    }
    if (t == 0) {
        float cnt0 = fmaxf(red[0][0] * red[2][0], 1.0f);
        float cnt1 = fmaxf(red[1][0] * red[3][0], 1.0f);
        out[0] = 0.5f * (acc[0] / cnt0 + acc[1] / cnt1) * 0.1f;
    }
}

__global__ void fape_zero(float* __restrict__ acc) {
    if (threadIdx.x < Bz) acc[threadIdx.x] = 0.0f;
}

// ---------------------------------------------------------------------------
extern "C" void kernel_launch(void* const* d_in, const int* in_sizes, int n_in,
                              void* d_out, int out_size, void* d_ws, size_t ws_size,
                              hipStream_t stream) {
    const float* pred          = (const float*)d_in[0];
    const float* truec         = (const float*)d_in[1];
    const unsigned char* cmask = (const unsigned char*)d_in[2];

    float* ws    = (float*)d_ws;
    float* F     = ws;                                   // Bz*Nn*KF  = 15360
    float* fmask = F + (size_t)Bz * Nn * KF;             // Bz*Nn     = 768
    float* P     = fmask + (size_t)Bz * Nn;              // Bz*NA*KF  = 215040
    float* pmask = P + (size_t)Bz * NA * KF;             // Bz*NA     = 10752
    float* acc   = pmask + (size_t)Bz * NA;              // Bz        = 2

    fape_zero<<<1, 32, 0, stream>>>(acc);
    fape_frames<<<(Bz * Nn + 255) / 256, 256, 0, stream>>>(pred, truec, cmask, F, fmask);
    fape_points<<<(Bz * NA + 255) / 256, 256, 0, stream>>>(pred, truec, cmask, P, pmask);
    fape_wmma<<<Bz * FT * SPLIT, 256, 0, stream>>>(F, P, fmask, pmask, acc);
    fape_finalize<<<1, 256, 0, stream>>>(fmask, pmask, acc, (float*)d_out);
}